// TrueHigherOrderAttention_39582418600238
// MI455X (gfx1250) — compile-verified
//
#include <hip/hip_runtime.h>
#include <cstdint>

typedef float v2f __attribute__((ext_vector_type(2)));
typedef float v8f __attribute__((ext_vector_type(8)));

namespace {
constexpr int Bn  = 4;
constexpr int Tn  = 128;
constexpr int Cn  = 512;
constexpr int NHn = 8;
constexpr int HSn = 64;
constexpr float SCALEF = 0.125f;  // 1/sqrt(64)
}

// -------------------------------------------------------------------------
// Dense row-major GEMM C[M,N] = A[M,K] @ B[K,N], fp32, WMMA 16x16x4.
// Block = 128 threads (4 waves) computing a 16x64 C panel (4 tiles).
// K is processed in chunks of 64 staged into LDS via CDNA5 async
// global->LDS loads (ASYNCcnt), then consumed by v_wmma_f32_16x16x4_f32.
// Requires: M%16==0, N%64==0, K%64==0, A/B 16-byte aligned rows.
// grid.x = (M/16) * (N/64)
// -------------------------------------------------------------------------
__global__ __launch_bounds__(128) void wmma_gemm_async(
    const float* __restrict__ A, const float* __restrict__ B,
    float* __restrict__ C, int M, int N, int K) {
  (void)M;
  __shared__ __align__(16) float As[16 * 64];   // 4 KB: A panel (16 rows x 64 k)
  __shared__ __align__(16) float Bs[64 * 64];   // 16 KB: B panel (64 k x 64 n)

  const int tid  = threadIdx.x;
  const int lane = tid & 31;
  const int wave = tid >> 5;
  const int half = lane >> 4;   // selects K pair within a 4-wide WMMA step
  const int l    = lane & 15;   // M (A frag) / N (B,C frag)

  const int nGroups = N >> 6;                   // number of 64-wide N panels
  const int tm = blockIdx.x / nGroups;          // 16-row panel index
  const int tg = blockIdx.x - tm * nGroups;     // 64-col panel index

  // Raw LDS byte offsets (low 32 bits of the generic LDS pointer).
  const unsigned asBase = (unsigned)(uintptr_t)(&As[0]);
  const unsigned bsBase = (unsigned)(uintptr_t)(&Bs[0]);

  v8f acc = {};
  for (int k0 = 0; k0 < K; k0 += 64) {
    // ---- stage A panel: 16x64 floats = 256 x 16B chunks (2 per thread) ----
#pragma unroll
    for (int c = tid; c < 256; c += 128) {
      const int row  = c >> 4;
      const int colf = (c & 15) << 2;
      const unsigned long long ga = (unsigned long long)(uintptr_t)(
          A + (size_t)(tm * 16 + row) * K + k0 + colf);
      asm volatile("global_load_async_to_lds_b128 %0, %1, off"
                   :: "v"(asBase + (unsigned)(c << 4)), "v"(ga)
                   : "memory");
    }
    // ---- stage B panel: 64x64 floats = 1024 x 16B chunks (8 per thread) ----
#pragma unroll
    for (int c = tid; c < 1024; c += 128) {
      const int row  = c >> 4;
      const int colf = (c & 15) << 2;
      const unsigned long long gb = (unsigned long long)(uintptr_t)(
          B + (size_t)(k0 + row) * N + tg * 64 + colf);
      asm volatile("global_load_async_to_lds_b128 %0, %1, off"
                   :: "v"(bsBase + (unsigned)(c << 4)), "v"(gb)
                   : "memory");
    }
    // Hint the next A panel into cache while this chunk computes.
    if (k0 + 64 < K)
      __builtin_prefetch(A + (size_t)(tm * 16 + l) * K + k0 + 64, 0, 3);

    asm volatile("s_wait_asynccnt 0" ::: "memory");
    __syncthreads();

    // ---- 16 WMMA steps from LDS ----
#pragma unroll
    for (int kk = 0; kk < 64; kk += 4) {
      const int kq = kk + (half << 1);
      v2f a = *(const v2f*)(&As[l * 64 + kq]);        // A[l][kq..kq+1]
      v2f b2;
      b2.x = Bs[kq * 64 + wave * 16 + l];             // B[kq][n]
      b2.y = Bs[(kq + 1) * 64 + wave * 16 + l];       // B[kq+1][n]
      acc = __builtin_amdgcn_wmma_f32_16x16x4_f32(
          false, a, false, b2, (short)0, acc, false, false);
    }
    __syncthreads();
  }

  const int tn = tg * 4 + wave;
  float* Cb = C + (size_t)(tm * 16 + (half << 3)) * N + tn * 16 + l;
#pragma unroll
  for (int r = 0; r < 8; ++r) Cb[(size_t)r * N] = acc[r];  // M = r + 8*half
}

// -------------------------------------------------------------------------
// att2[bh,i,j] = SCALE * sum_d p0[bh,i,d] * p1[bh,j,d]
// P0/P1 stored as (B*T, C) with head columns at h*HS. Per z=(b*NH+h):
// M=N=T=128, K=HS=64. 64 tiles/z -> 16 blocks of 4 waves. grid=(16, B*NH).
// -------------------------------------------------------------------------
__global__ __launch_bounds__(128) void attn_scores_kernel(
    const float* __restrict__ P0, const float* __restrict__ P1,
    float* __restrict__ ATT) {
  const int z = blockIdx.y;
  const int b = z / NHn, h = z - b * NHn;
  const int lane = threadIdx.x & 31;
  const int wave = threadIdx.x >> 5;
  const int half = lane >> 4;
  const int l    = lane & 15;
  const int tile = blockIdx.x * 4 + wave;      // 0..63
  const int tm = tile >> 3, tn = tile & 7;

  const float* Ab = P0 + (size_t)(b * Tn + tm * 16 + l) * Cn + h * HSn;
  const float* Bb = P1 + (size_t)(b * Tn + tn * 16 + l) * Cn + h * HSn;
  v8f acc = {};
#pragma unroll
  for (int k0 = 0; k0 < HSn; k0 += 4) {
    const int kk = k0 + (half << 1);
    v2f a  = *(const v2f*)(Ab + kk);   // p0[i=l][kk..kk+1]
    v2f b2 = *(const v2f*)(Bb + kk);   // p1[j=l][kk..kk+1] == (p1^T)[kk][l]
    acc = __builtin_amdgcn_wmma_f32_16x16x4_f32(
        false, a, false, b2, (short)0, acc, false, false);
  }
  float* Cb = ATT + (size_t)z * Tn * Tn +
              (size_t)(tm * 16 + (half << 3)) * Tn + tn * 16 + l;
#pragma unroll
  for (int r = 0; r < 8; ++r) Cb[(size_t)r * Tn] = acc[r] * SCALEF;
}

// -------------------------------------------------------------------------
// Per-head value projection: out[z,t,:] = P[(b*T+t), h*HS + :] @ W (HSxHS).
// Per z: M=T=128, N=HS=64, K=HS=64 -> 32 tiles -> 8 blocks. grid=(8, B*NH).
// -------------------------------------------------------------------------
__global__ __launch_bounds__(128) void head_gemm_kernel(
    const float* __restrict__ P, const float* __restrict__ W,
    float* __restrict__ OUT) {
  const int z = blockIdx.y;
  const int b = z / NHn, h = z - b * NHn;
  const int lane = threadIdx.x & 31;
  const int wave = threadIdx.x >> 5;
  const int half = lane >> 4;
  const int l    = lane & 15;
  const int tile = blockIdx.x * 4 + wave;      // 0..31
  const int tm = tile >> 2, tn = tile & 3;

  const float* Ab = P + (size_t)(b * Tn + tm * 16 + l) * Cn + h * HSn;
  const float* Bb = W + tn * 16 + l;
  v8f acc = {};
#pragma unroll
  for (int k0 = 0; k0 < HSn; k0 += 4) {
    const int kk = k0 + (half << 1);
    v2f a = *(const v2f*)(Ab + kk);
    v2f b2;
    b2.x = Bb[(size_t)kk * HSn];
    b2.y = Bb[(size_t)(kk + 1) * HSn];
    acc = __builtin_amdgcn_wmma_f32_16x16x4_f32(
        false, a, false, b2, (short)0, acc, false, false);
  }
  float* Cb = OUT + (size_t)z * Tn * HSn +
              (size_t)(tm * 16 + (half << 3)) * HSn + tn * 16 + l;
#pragma unroll
  for (int r = 0; r < 8; ++r) Cb[(size_t)r * HSn] = acc[r];
}

// s2[bh*T + t] = sum_d p2[bh, t, d]
__global__ __launch_bounds__(256) void rowsum_kernel(
    const float* __restrict__ P2, float* __restrict__ S2) {
  const int idx = blockIdx.x * 256 + threadIdx.x;  // < B*NH*T
  const int z = idx / Tn, t = idx - z * Tn;
  const int b = z / NHn, h = z - b * NHn;
  const float* p = P2 + (size_t)(b * Tn + t) * Cn + h * HSn;
  float s = 0.f;
#pragma unroll 8
  for (int d = 0; d < HSn; ++d) s += p[d];
  S2[idx] = s;
}

// -------------------------------------------------------------------------
// Core higher-order softmax + weighted sum. One workgroup per (b,h,i).
// y[i,d] = (1/Z) sum_{j<=i} V0h[j,d] * sum_{k<=j} exp(a[i,j]*s[k]-m) * V1h[k,d]
// 8 waves split j; 32 lanes split d (2 floats each). lz is lane-uniform.
// -------------------------------------------------------------------------
__global__ __launch_bounds__(256) void hoa_kernel(
    const float* __restrict__ ATT, const float* __restrict__ S2,
    const float* __restrict__ V0h, const float* __restrict__ V1h,
    float* __restrict__ Y) {
  __shared__ float sa[Tn];
  __shared__ float ss[Tn];
  __shared__ float rbuf[256];
  __shared__ float ybuf[8][HSn];

  const int gid = blockIdx.x;          // = bh*T + i
  const int i  = gid & (Tn - 1);
  const int bh = gid >> 7;             // Tn = 128
  const int tid  = threadIdx.x;
  const int wave = tid >> 5;
  const int lane = tid & 31;

  const float* arow = ATT + (size_t)bh * Tn * Tn + (size_t)i * Tn;
  if (tid < Tn) {
    sa[tid] = arow[tid];
    ss[tid] = S2[bh * Tn + tid];
  }
  __syncthreads();

  // pass 1: masked row max over {(j,k): j<=i, k<=j}
  float lm = -3.4e38f;
  for (int j = tid; j <= i; j += 256) {
    const float a = sa[j];
    for (int k = 0; k <= j; ++k) lm = fmaxf(lm, a * ss[k]);
  }
  rbuf[tid] = lm;
  __syncthreads();
  for (int s = 128; s > 0; s >>= 1) {
    if (tid < s) rbuf[tid] = fmaxf(rbuf[tid], rbuf[tid + s]);
    __syncthreads();
  }
  const float m = rbuf[0];
  __syncthreads();  // rbuf reused below

  // pass 2: weighted accumulation + partition function
  const float* V0b = V0h + (size_t)bh * Tn * HSn;
  const float* V1b = V1h + (size_t)bh * Tn * HSn;
  const int d = lane * 2;
  float accx = 0.f, accy = 0.f, lz = 0.f;
  for (int j = wave; j <= i; j += 8) {
    const float a = sa[j];
    float wx = 0.f, wy = 0.f;
    const float* v1 = V1b + d;
    for (int k = 0; k <= j; ++k) {
      const float e = __expf(a * ss[k] - m);
      lz += e;
      wx += e * v1[0];
      wy += e * v1[1];
      v1 += HSn;
    }
    const float* v0 = V0b + (size_t)j * HSn + d;
    accx += v0[0] * wx;
    accy += v0[1] * wy;
  }
  ybuf[wave][d]     = accx;
  ybuf[wave][d + 1] = accy;
  if (lane == 0) rbuf[wave] = lz;   // lz is identical across lanes of a wave
  __syncthreads();

  float Z = 0.f;
#pragma unroll
  for (int w = 0; w < 8; ++w) Z += rbuf[w];

  if (tid < HSn) {
    float v = 0.f;
#pragma unroll
    for (int w = 0; w < 8; ++w) v += ybuf[w][tid];
    const int b = bh / NHn, h = bh - b * NHn;
    Y[(size_t)(b * Tn + i) * Cn + h * HSn + tid] = v / Z;
  }
}

// -------------------------------------------------------------------------
extern "C" void kernel_launch(void* const* d_in, const int* in_sizes, int n_in,
                              void* d_out, int out_size, void* d_ws, size_t ws_size,
                              hipStream_t stream) {
  (void)in_sizes; (void)n_in; (void)out_size; (void)ws_size;
  const float* x   = (const float*)d_in[0];
  const float* Wp0 = (const float*)d_in[1];
  const float* Wp1 = (const float*)d_in[2];
  const float* Wp2 = (const float*)d_in[3];
  const float* Wv0 = (const float*)d_in[4];
  const float* Wv1 = (const float*)d_in[5];
  const float* Wc  = (const float*)d_in[6];
  float* out = (float*)d_out;

  const size_t BTC  = (size_t)Bn * Tn * Cn;        // 262144
  const size_t BHTT = (size_t)Bn * NHn * Tn * Tn;  // 524288
  const size_t BHT  = (size_t)Bn * NHn * Tn;       // 4096
  const size_t BHTH = (size_t)Bn * NHn * Tn * HSn; // 262144

  float* P0  = (float*)d_ws;
  float* P1  = P0 + BTC;
  float* P2  = P1 + BTC;
  float* ATT = P2 + BTC;
  float* S2  = ATT + BHTT;
  float* V0h = S2 + BHT;
  float* V1h = V0h + BHTH;
  float* Yb  = V1h + BHTH;

  const int M = Bn * Tn;                          // 512
  const int projBlocks = (M / 16) * (Cn / 64);    // 32 * 8 = 256

  // Projections: p0, p1, p2 = x @ Wp{0,1,2}   (512x512x512, async-LDS + WMMA)
  wmma_gemm_async<<<projBlocks, 128, 0, stream>>>(x, Wp0, P0, M, Cn, Cn);
  wmma_gemm_async<<<projBlocks, 128, 0, stream>>>(x, Wp1, P1, M, Cn, Cn);
  wmma_gemm_async<<<projBlocks, 128, 0, stream>>>(x, Wp2, P2, M, Cn, Cn);

  // s2[bh,k] = sum_d p2[bh,k,d]
  rowsum_kernel<<<(Bn * NHn * Tn) / 256, 256, 0, stream>>>(P2, S2);

  // Value projections per head (shared HSxHS weights)
  head_gemm_kernel<<<dim3(8, Bn * NHn), 128, 0, stream>>>(P1, Wv0, V0h);
  head_gemm_kernel<<<dim3(8, Bn * NHn), 128, 0, stream>>>(P2, Wv1, V1h);

  // Pairwise scores att2 = SCALE * p0 p1^T
  attn_scores_kernel<<<dim3(16, Bn * NHn), 128, 0, stream>>>(P0, P1, ATT);

  // Higher-order masked softmax + value contraction
  hoa_kernel<<<Bn * NHn * Tn, 256, 0, stream>>>(ATT, S2, V0h, V1h, Yb);

  // Output projection
  wmma_gemm_async<<<projBlocks, 128, 0, stream>>>(Yb, Wc, out, M, Cn, Cn);
}